// CAREConv_66099546685570
// MI455X (gfx1250) — compile-verified
//
#include <hip/hip_runtime.h>
#include <math.h>

#define K_DEG     32
#define IN_FEATS  128
#define H_FEATS   64
#define K_SEL     16
#define LDS_STRIDE (IN_FEATS + 4)   // pad to kill LDS bank conflicts on b64 reads

typedef __attribute__((ext_vector_type(2))) float v2f;
typedef __attribute__((ext_vector_type(8))) float v8f;

// ---------------------------------------------------------------------------
// Kernel 1: z[i][c] = tanh(dot(x[i,:], W_mlp[c,:]) + b_mlp[c]),  c in {0,1}
// One wave per node. Lane covers features [4*lane, 4*lane+4).
// ---------------------------------------------------------------------------
__global__ void __launch_bounds__(256)
care_edge_embed(const float* __restrict__ x,
                const float* __restrict__ Wm,
                const float* __restrict__ bm,
                float* __restrict__ z, int n)
{
    const int lane = threadIdx.x & 31;
    const int wave = threadIdx.x >> 5;
    const int node = blockIdx.x * 8 + wave;
    if (node >= n) return;

    const float4 xv = *(const float4*)(x + (size_t)node * IN_FEATS + lane * 4);
    const float4 w0 = *(const float4*)(Wm + lane * 4);
    const float4 w1 = *(const float4*)(Wm + IN_FEATS + lane * 4);

    float p0 = xv.x * w0.x + xv.y * w0.y + xv.z * w0.z + xv.w * w0.w;
    float p1 = xv.x * w1.x + xv.y * w1.y + xv.z * w1.z + xv.w * w1.w;
    #pragma unroll
    for (int off = 16; off > 0; off >>= 1) {
        p0 += __shfl_xor(p0, off, 32);
        p1 += __shfl_xor(p1, off, 32);
    }
    if (lane == 0) {
        z[node * 2 + 0] = tanhf(p0 + bm[0]);
        z[node * 2 + 1] = tanhf(p1 + bm[1]);
    }
}

// ---------------------------------------------------------------------------
// Kernel 2: per-node edge distance, stable top-16 selection per relation,
// mean-aggregate x over selected sources, residual + 0.5 weighting:
//   h[i] = x[i] + (0.5/16) * sum_r sum_{sel} x[src_r_sel]
// One wave per node; lane k owns edge k (K=32 == wave32).
// Selection: rank_k = #{j : d_j < d_k || (d_j == d_k && j < k)}; keep rank<16
// (exactly jax.lax.top_k(-d) semantics; exactly 16 lanes selected).
// ---------------------------------------------------------------------------
__global__ void __launch_bounds__(256)
care_aggregate(const float* __restrict__ x,
               const int* __restrict__ src0,
               const int* __restrict__ src1,
               const int* __restrict__ src2,
               const float* __restrict__ z,
               float* __restrict__ h, int n)
{
    const int lane = threadIdx.x & 31;
    const int wave = threadIdx.x >> 5;
    const int node = blockIdx.x * 8 + wave;
    if (node >= n) return;

    const float zi0 = z[node * 2 + 0];
    const float zi1 = z[node * 2 + 1];

    float4 acc = make_float4(0.f, 0.f, 0.f, 0.f);
    const int* __restrict__ srcs[3] = { src0, src1, src2 };

    for (int r = 0; r < 3; ++r) {
        const int s = srcs[r][(size_t)node * K_DEG + lane];
        const float d = fabsf(z[s * 2 + 0] - zi0) + fabsf(z[s * 2 + 1] - zi1);

        int rank = 0;
        #pragma unroll
        for (int j = 0; j < 32; ++j) {
            const float dj = __shfl(d, j, 32);
            rank += (dj < d) || (dj == d && j < lane) ? 1 : 0;
        }
        // wave-uniform mask of the 16 winners
        unsigned mask = (unsigned)__ballot(rank < K_SEL);
        #pragma unroll
        for (int t = 0; t < K_SEL; ++t) {
            const int l = __ffs(mask) - 1;     // uniform across lanes
            mask &= mask - 1;
            const int ssel = __shfl(s, l, 32);
            const float4 xv =
                *(const float4*)(x + (size_t)ssel * IN_FEATS + lane * 4);
            acc.x += xv.x; acc.y += xv.y; acc.z += xv.z; acc.w += xv.w;
        }
    }

    const float4 xi = *(const float4*)(x + (size_t)node * IN_FEATS + lane * 4);
    const float sc = 0.5f / (float)K_SEL;      // P / K_SEL
    float4 hv;
    hv.x = fmaf(sc, acc.x, xi.x);
    hv.y = fmaf(sc, acc.y, xi.y);
    hv.z = fmaf(sc, acc.z, xi.z);
    hv.w = fmaf(sc, acc.w, xi.w);
    *(float4*)(h + (size_t)node * IN_FEATS + lane * 4) = hv;
}

// ---------------------------------------------------------------------------
// Kernel 3: out = h @ W_lin^T + b_lin  via V_WMMA_F32_16X16X4_F32 (pure fp32).
// Block = 256 (8 waves). W_lin (64x128, 32KB) staged in LDS with +4 row pad.
// Wave handles one 16-row node tile; 4 N-tiles x 32 K-steps = 128 WMMAs.
// VGPR layouts (ISA 7.12.2): lane ln, g=ln>>4, m/n=ln&15:
//   A vgpr j : A[m][K = 2g + j]      -> contiguous float2 at h[row*128 + kk + 2g]
//   B vgpr j : B[K = 2g + j][n]      -> contiguous float2 at Wl[n*128 + kk + 2g]
//   C/D vgpr j : C[m = j + 8g][n]
// ---------------------------------------------------------------------------
__global__ void __launch_bounds__(256)
care_out_gemm(const float* __restrict__ h,
              const float* __restrict__ Wl,
              const float* __restrict__ bl,
              float* __restrict__ out, int n)
{
    __shared__ __align__(16) float lds_w[H_FEATS * LDS_STRIDE];

    const int tid = threadIdx.x;
    // cooperative padded copy of W_lin into LDS (float4 granularity)
    for (int i4 = tid; i4 < (H_FEATS * IN_FEATS) / 4; i4 += 256) {
        const int o  = i4 >> 5;            // i4 / (IN_FEATS/4)
        const int k4 = (i4 & 31) * 4;
        *(float4*)(lds_w + o * LDS_STRIDE + k4) = ((const float4*)Wl)[i4];
    }
    __syncthreads();

    const int lane = tid & 31;
    const int wave = tid >> 5;
    const int g    = lane >> 4;
    const int ln   = lane & 15;

    const int tile = blockIdx.x * 8 + wave;
    const int row0 = tile * 16;
    if (row0 >= n) return;

    const int arow = min(row0 + ln, n - 1);
    const float* __restrict__ aptr = h + (size_t)arow * IN_FEATS + 2 * g;

    v8f c0 = {}, c1 = {}, c2 = {}, c3 = {};

    #pragma unroll 4
    for (int kk = 0; kk < IN_FEATS; kk += 4) {
        const v2f a  = *(const v2f*)(aptr + kk);
        const v2f b0 = *(const v2f*)(lds_w + ( 0 + ln) * LDS_STRIDE + kk + 2 * g);
        const v2f b1 = *(const v2f*)(lds_w + (16 + ln) * LDS_STRIDE + kk + 2 * g);
        const v2f b2 = *(const v2f*)(lds_w + (32 + ln) * LDS_STRIDE + kk + 2 * g);
        const v2f b3 = *(const v2f*)(lds_w + (48 + ln) * LDS_STRIDE + kk + 2 * g);
        c0 = __builtin_amdgcn_wmma_f32_16x16x4_f32(false, a, false, b0, (short)0, c0, false, false);
        c1 = __builtin_amdgcn_wmma_f32_16x16x4_f32(false, a, false, b1, (short)0, c1, false, false);
        c2 = __builtin_amdgcn_wmma_f32_16x16x4_f32(false, a, false, b2, (short)0, c2, false, false);
        c3 = __builtin_amdgcn_wmma_f32_16x16x4_f32(false, a, false, b3, (short)0, c3, false, false);
    }

    v8f cs[4] = { c0, c1, c2, c3 };
    const int rbase = row0 + 8 * g;
    if (row0 + 16 <= n) {
        // Fast path (always taken when n % 16 == 0): no per-store exec masking.
        #pragma unroll
        for (int nt = 0; nt < 4; ++nt) {
            const float bb = bl[nt * 16 + ln];
            float* __restrict__ optr = out + (size_t)rbase * H_FEATS + nt * 16 + ln;
            #pragma unroll
            for (int j = 0; j < 8; ++j)
                optr[(size_t)j * H_FEATS] = cs[nt][j] + bb;
        }
    } else {
        #pragma unroll
        for (int nt = 0; nt < 4; ++nt) {
            const float bb = bl[nt * 16 + ln];
            #pragma unroll
            for (int j = 0; j < 8; ++j) {
                const int row = rbase + j;
                if (row < n)
                    out[(size_t)row * H_FEATS + nt * 16 + ln] = cs[nt][j] + bb;
            }
        }
    }
}

// ---------------------------------------------------------------------------
// Launch
// inputs: 0:x [N,128] 1:src0 [N,32] 2:src1 3:src2 4:W_mlp [2,128]
//         5:b_mlp [2]  6:W_lin [64,128]  7:b_lin [64]      out: [N,64]
// ---------------------------------------------------------------------------
extern "C" void kernel_launch(void* const* d_in, const int* in_sizes, int n_in,
                              void* d_out, int out_size, void* d_ws, size_t ws_size,
                              hipStream_t stream) {
    const float* x    = (const float*)d_in[0];
    const int*   src0 = (const int*)d_in[1];
    const int*   src1 = (const int*)d_in[2];
    const int*   src2 = (const int*)d_in[3];
    const float* Wm   = (const float*)d_in[4];
    const float* bm   = (const float*)d_in[5];
    const float* Wl   = (const float*)d_in[6];
    const float* bl   = (const float*)d_in[7];
    float* out = (float*)d_out;

    const int n = in_sizes[0] / IN_FEATS;    // N = 50000

    float* z = (float*)d_ws;                 // [N,2]
    float* h = z + (size_t)n * 2;            // [N,128]

    const int node_blocks = (n + 7) / 8;     // 8 waves/block, wave per node
    care_edge_embed<<<node_blocks, 256, 0, stream>>>(x, Wm, bm, z, n);
    care_aggregate<<<node_blocks, 256, 0, stream>>>(x, src0, src1, src2, z, h, n);

    const int tiles = (n + 15) / 16;
    const int gemm_blocks = (tiles + 7) / 8; // 8 waves/block, wave per 16-row tile
    care_out_gemm<<<gemm_blocks, 256, 0, stream>>>(h, Wl, bl, out, n);
}